// SGC_5686536700273
// MI455X (gfx1250) — compile-verified
//
#include <hip/hip_runtime.h>
#include <hip/hip_bf16.h>

typedef __attribute__((ext_vector_type(2))) float v2f;
typedef __attribute__((ext_vector_type(8))) float v8f;

// ---------------------------------------------------------------------------
// GCN normalization helpers
// ---------------------------------------------------------------------------
__global__ void set_ones_kernel(float* __restrict__ p, int n) {
    int i = blockIdx.x * blockDim.x + threadIdx.x;
    if (i < n) p[i] = 1.0f;  // self-loop contributes 1 to every node's degree
}

__global__ void deg_scatter_kernel(const long long* __restrict__ dst,
                                   float* __restrict__ deg, int nE) {
    int e = blockIdx.x * blockDim.x + threadIdx.x;
    if (e < nE) atomicAdd(&deg[(int)dst[e]], 1.0f);
}

__global__ void rsqrt_kernel(float* __restrict__ p, int n) {
    int i = blockIdx.x * blockDim.x + threadIdx.x;
    if (i < n) {
        float d = p[i];
        p[i] = d > 0.0f ? rsqrtf(d) : 0.0f;
    }
}

__global__ void edge_norm_kernel(const long long* __restrict__ src,
                                 const long long* __restrict__ dst,
                                 const float* __restrict__ dinv,
                                 float* __restrict__ norm, int nE) {
    int e = blockIdx.x * blockDim.x + threadIdx.x;
    if (e < nE) norm[e] = dinv[(int)src[e]] * dinv[(int)dst[e]];
}

// ---------------------------------------------------------------------------
// Propagation: y = A_hat x   =>   y[i] = dinv[i]^2 * x[i]  (self loop, init)
//                                 y[dst] += norm_e * x[src] (edge scatter)
// ---------------------------------------------------------------------------
template <int F>
__global__ void selfloop_init_kernel(const float* __restrict__ x,
                                     const float* __restrict__ dinv,
                                     float* __restrict__ y, int n) {
    constexpr int FQ = F / 4;
    int idx = blockIdx.x * blockDim.x + threadIdx.x;          // over n*F/4
    if (idx >= n * FQ) return;
    int i = idx / FQ;
    float s = dinv[i];
    s *= s;
    float4 v = ((const float4*)x)[idx];
    v.x *= s; v.y *= s; v.z *= s; v.w *= s;
    ((float4*)y)[idx] = v;
}

template <int F>
__global__ void edge_scatter_kernel(const long long* __restrict__ src,
                                    const long long* __restrict__ dst,
                                    const float* __restrict__ norm,
                                    const float* __restrict__ x,
                                    float* __restrict__ y, int nE) {
    constexpr int TPE = F / 4;        // threads per edge (float4 each)
    constexpr int EPB = 256 / TPE;    // edges per 256-thread block
    int t = threadIdx.x;
    int e = blockIdx.x * EPB + t / TPE;
    if (e >= nE) return;
    int f4 = (t % TPE) * 4;
    long long s = src[e];
    long long d = dst[e];
    float w = norm[e];
    float4 v = *(const float4*)(x + s * F + f4);
    float* yp = y + d * F + f4;
    atomicAdd(yp + 0, v.x * w);
    atomicAdd(yp + 1, v.y * w);
    atomicAdd(yp + 2, v.z * w);
    atomicAdd(yp + 3, v.w * w);
}

template <int F, bool RELU>
__global__ void bias_act_kernel(const float* __restrict__ in,
                                const float* __restrict__ bias,
                                float* __restrict__ out, int n) {
    int idx = blockIdx.x * blockDim.x + threadIdx.x;  // over n*F
    if (idx >= n * F) return;
    float r = in[idx] + bias[idx % F];
    if (RELU) r = fmaxf(r, 0.0f);
    out[idx] = r;
}

// ---------------------------------------------------------------------------
// WMMA f32 GEMM:  Y[n x NCOL] = A[n x K] * W[K x NCOL] (+bias) (+relu)
// One wave -> one 16-row tile, NT = NCOLP/16 column tiles held in registers.
// Uses V_WMMA_F32_16X16X4_F32 (wave32, f32 in/out -> matches reference math).
// ---------------------------------------------------------------------------
template <int K, int NCOL, int NCOLP, bool RELU, bool HASBIAS>
__global__ void __launch_bounds__(256)
wmma_gemm_kernel(const float* __restrict__ A, const float* __restrict__ W,
                 const float* __restrict__ bias, float* __restrict__ Y,
                 int nrows) {
    constexpr int NT = NCOLP / 16;
    __shared__ float sW[K * NCOLP];
    __shared__ float sB[NCOLP];

    int tid = threadIdx.x;
    for (int idx = tid; idx < K * NCOLP; idx += 256) {
        int r = idx / NCOLP, c = idx % NCOLP;
        sW[idx] = (c < NCOL) ? W[r * NCOL + c] : 0.0f;
    }
    if (tid < NCOLP) sB[tid] = (HASBIAS && tid < NCOL) ? bias[tid] : 0.0f;
    __syncthreads();

    int wave = tid >> 5;
    int lane = tid & 31;
    int tile = blockIdx.x * 8 + wave;
    int ntiles = (nrows + 15) >> 4;
    if (tile >= ntiles) return;          // uniform per wave

    int rowBase = tile * 16;
    int half = lane >> 4;                // 0: lanes 0-15, 1: lanes 16-31
    int l15 = lane & 15;
    int koff = half * 2;                 // A/B VGPR halves carry K+2,K+3

    int arow = rowBase + l15;
    if (arow >= nrows) arow = nrows - 1; // clamp (never hit: 16 | 50000)
    const float* Arow = A + (long long)arow * K;

    v8f acc[NT];
#pragma unroll
    for (int t = 0; t < NT; ++t)
        acc[t] = (v8f){0.f, 0.f, 0.f, 0.f, 0.f, 0.f, 0.f, 0.f};

#pragma unroll 4
    for (int k = 0; k < K; k += 4) {
        // A 16x4 tile: lane l15 = row, {x,y} = K columns (k+koff, k+koff+1)
        v2f a = *(const v2f*)(Arow + k + koff);   // 8B aligned: k%4==0, koff even
#pragma unroll
        for (int t = 0; t < NT; ++t) {
            v2f b;
            b.x = sW[(k + koff) * NCOLP + t * 16 + l15];
            b.y = sW[(k + koff + 1) * NCOLP + t * 16 + l15];
            acc[t] = __builtin_amdgcn_wmma_f32_16x16x4_f32(
                false, a, false, b, (short)0, acc[t], false, false);
        }
    }

    // C/D layout: VGPR v -> M = v + 8*half, N = l15
    int mOff = half * 8;
#pragma unroll
    for (int t = 0; t < NT; ++t) {
#pragma unroll
        for (int v = 0; v < 8; ++v) {
            int c = t * 16 + l15;
            int rowOut = rowBase + v + mOff;
            float r = acc[t][v] + sB[t * 16 + l15];
            if (RELU) r = fmaxf(r, 0.0f);
            if ((NCOL == NCOLP || c < NCOL) && rowOut < nrows)
                Y[(long long)rowOut * NCOL + c] = r;
        }
    }
}

// ---------------------------------------------------------------------------
// Launch. Inputs: x[N*128], edge_index[2*E] (int64), W1,b1,W2,b2,Wf,bf.
// Output: embedding[N*32] ++ logits[N*40].
// Key rewrite: A^2 (x W1) == (A^2 x) W1  -> GEMM first, propagate narrow.
// ---------------------------------------------------------------------------
extern "C" void kernel_launch(void* const* d_in, const int* in_sizes, int n_in,
                              void* d_out, int out_size, void* d_ws, size_t ws_size,
                              hipStream_t stream) {
    const float*     x   = (const float*)d_in[0];
    const long long* ei  = (const long long*)d_in[1];
    const float*     W1  = (const float*)d_in[2];
    const float*     b1  = (const float*)d_in[3];
    const float*     W2  = (const float*)d_in[4];
    const float*     b2  = (const float*)d_in[5];
    const float*     Wf  = (const float*)d_in[6];
    const float*     bf  = (const float*)d_in[7];

    const int nN = in_sizes[0] / 128;   // 50000
    const int nE = in_sizes[1] / 2;     // 800000
    const long long* src = ei;
    const long long* dst = ei + nE;

    float* ws   = (float*)d_ws;
    float* dinv = ws;                   // [nN]
    float* norm = dinv + nN;            // [nE]
    float* bufA = norm + nE;            // [nN*64]
    float* bufB = bufA + (size_t)nN * 64;  // [nN*64]

    float* embed  = (float*)d_out;                     // [nN*32]
    float* logits = (float*)d_out + (size_t)nN * 32;   // [nN*40]

    dim3 blk(256);
    auto cdiv = [](int a, int b) { return (a + b - 1) / b; };

    // --- GCN norm -----------------------------------------------------------
    set_ones_kernel<<<cdiv(nN, 256), blk, 0, stream>>>(dinv, nN);
    deg_scatter_kernel<<<cdiv(nE, 256), blk, 0, stream>>>(dst, dinv, nE);
    rsqrt_kernel<<<cdiv(nN, 256), blk, 0, stream>>>(dinv, nN);
    edge_norm_kernel<<<cdiv(nE, 256), blk, 0, stream>>>(src, dst, dinv, norm, nE);

    const int gemmBlocks = cdiv(cdiv(nN, 16), 8);

    // --- Layer 1: g = x @ W1 (128->64), then 2 hops at 64-wide, then +b1,relu
    wmma_gemm_kernel<128, 64, 64, false, false>
        <<<gemmBlocks, blk, 0, stream>>>(x, W1, nullptr, bufA, nN);
    selfloop_init_kernel<64><<<cdiv(nN * 16, 256), blk, 0, stream>>>(bufA, dinv, bufB, nN);
    edge_scatter_kernel<64><<<cdiv(nE, 16), blk, 0, stream>>>(src, dst, norm, bufA, bufB, nE);
    selfloop_init_kernel<64><<<cdiv(nN * 16, 256), blk, 0, stream>>>(bufB, dinv, bufA, nN);
    edge_scatter_kernel<64><<<cdiv(nE, 16), blk, 0, stream>>>(src, dst, norm, bufB, bufA, nE);
    bias_act_kernel<64, true><<<cdiv(nN * 64, 256), blk, 0, stream>>>(bufA, b1, bufA, nN);

    // --- Layer 2: g = h @ W2 (64->32), then 2 hops at 32-wide, then +b2,relu
    wmma_gemm_kernel<64, 32, 32, false, false>
        <<<gemmBlocks, blk, 0, stream>>>(bufA, W2, nullptr, bufB, nN);
    selfloop_init_kernel<32><<<cdiv(nN * 8, 256), blk, 0, stream>>>(bufB, dinv, bufA, nN);
    edge_scatter_kernel<32><<<cdiv(nE, 32), blk, 0, stream>>>(src, dst, norm, bufB, bufA, nE);
    selfloop_init_kernel<32><<<cdiv(nN * 8, 256), blk, 0, stream>>>(bufA, dinv, bufB, nN);
    edge_scatter_kernel<32><<<cdiv(nE, 32), blk, 0, stream>>>(src, dst, norm, bufA, bufB, nE);
    bias_act_kernel<32, true><<<cdiv(nN * 32, 256), blk, 0, stream>>>(bufB, b2, embed, nN);

    // --- Head: logits = embed @ Wf + bf (32->40, pad cols to 48) ------------
    wmma_gemm_kernel<32, 40, 48, false, true>
        <<<gemmBlocks, blk, 0, stream>>>(embed, Wf, bf, logits, nN);
}